// Detect_67310727463192
// MI455X (gfx1250) — compile-verified
//
#include <hip/hip_runtime.h>
#include <cstdint>

#define NUM_IMG 16
#define PRIORS  131072
#define K_SEL   5000
#define TOPK    750
#define CONF_TH 0.05f
#define NMS_TH  0.3f
#define NEGF    (-1e9f)
#define CHUNK   5            // ceil(5000 / 1024) register-resident entries/thread

// Monotonic float -> uint key (larger float => larger key)
__device__ __forceinline__ unsigned keyOf(float f) {
    unsigned u = __float_as_uint(f);
    return (u & 0x80000000u) ? ~u : (u | 0x80000000u);
}

// ---------------------------------------------------------------------------
// init: zero class-0 output plane, reset counters / radix state / selection
// ---------------------------------------------------------------------------
__global__ void init_kernel(float* __restrict__ out,
                            float* __restrict__ selScores,
                            float4* __restrict__ selBoxes,
                            unsigned* __restrict__ prefix,
                            int* __restrict__ kRem,
                            int* __restrict__ cnt) {
    int j = blockIdx.x * blockDim.x + threadIdx.x;
    if (j < NUM_IMG * K_SEL) {
        selScores[j] = NEGF;
        selBoxes[j]  = make_float4(0.f, 0.f, 0.f, 0.f);
    }
    if (j < NUM_IMG * TOPK * 5) {
        int img = j / (TOPK * 5);
        int rem = j % (TOPK * 5);
        out[(size_t)img * 2 * TOPK * 5 + rem] = 0.0f;   // background class plane
    }
    if (j < NUM_IMG) { prefix[j] = 0u; kRem[j] = K_SEL; cnt[j] = 0; }
}

// ---------------------------------------------------------------------------
// radix-select pass: 256-bin histogram of key byte `pass`, restricted to keys
// matching the prefix selected by previous passes. One workgroup per image.
// ---------------------------------------------------------------------------
__global__ __launch_bounds__(1024)
void hist_kernel(const float2* __restrict__ conf2,
                 unsigned* __restrict__ hist,
                 const unsigned* __restrict__ prefix, int pass) {
    __shared__ int lh[256];
    const int img = blockIdx.x;
    const int tid = threadIdx.x;
    for (int i = tid; i < 256; i += 1024) lh[i] = 0;
    __syncthreads();
    const unsigned pre    = prefix[img];
    const unsigned maskHi = (pass == 0) ? 0u : (0xFFFFFFFFu << (32 - 8 * pass));
    const int shift = 24 - 8 * pass;
    const float2* base = conf2 + (size_t)img * PRIORS;
    for (int p = tid; p < PRIORS; p += 1024) {
        __builtin_prefetch(base + p + 4096, 0, 1);     // global_prefetch_b8
        float c = base[p].y;                            // class-1 score
        float s = (c > CONF_TH) ? c : NEGF;
        unsigned k = keyOf(s);
        if (((k ^ pre) & maskHi) == 0u)
            atomicAdd(&lh[(k >> shift) & 255u], 1);
    }
    __syncthreads();
    for (int i = tid; i < 256; i += 1024) hist[img * 256 + i] = (unsigned)lh[i];
}

__global__ void scan_kernel(const unsigned* __restrict__ hist,
                            unsigned* __restrict__ prefix,
                            int* __restrict__ kRem, int pass) {
    int img = threadIdx.x;
    if (img >= NUM_IMG) return;
    const int shift = 24 - 8 * pass;
    int k = kRem[img];
    int cum = 0, b = 0;
    for (int i = 255; i >= 0; --i) {
        int c = (int)hist[img * 256 + i];
        if (cum + c >= k) { b = i; break; }
        cum += c;
    }
    prefix[img] |= ((unsigned)b) << shift;
    kRem[img]    = k - cum;
}

// ---------------------------------------------------------------------------
// compaction: gather survivors (key > T, then key == T up to K_SEL), decoding
// the SSD box only for the selected priors.
// ---------------------------------------------------------------------------
__global__ void compact_kernel(const float4* __restrict__ loc4,
                               const float2* __restrict__ conf2,
                               const float4* __restrict__ prior4,
                               const unsigned* __restrict__ prefixT,
                               int* __restrict__ cnt,
                               float* __restrict__ selScores,
                               float4* __restrict__ selBoxes, int eqPass) {
    int g = blockIdx.x * blockDim.x + threadIdx.x;
    if (g >= NUM_IMG * PRIORS) return;
    const int img = g / PRIORS, p = g % PRIORS;
    float c = conf2[g].y;
    float s = (c > CONF_TH) ? c : NEGF;
    unsigned k = keyOf(s);
    unsigned T = prefixT[img];
    bool sel = eqPass ? (k == T) : (k > T);
    if (!sel) return;
    int pos = atomicAdd(&cnt[img], 1);
    if (pos >= K_SEL) return;
    float4 l  = loc4[g];
    float4 pr = prior4[p];
    float cx = pr.x + l.x * 0.1f * pr.z;
    float cy = pr.y + l.y * 0.1f * pr.w;
    float w  = pr.z * expf(l.z * 0.2f);
    float h  = pr.w * expf(l.w * 0.2f);
    selScores[img * K_SEL + pos] = s;
    selBoxes [img * K_SEL + pos] =
        make_float4(cx - 0.5f * w, cy - 0.5f * h, cx + 0.5f * w, cy + 0.5f * h);
}

// ---------------------------------------------------------------------------
// NMS: one workgroup (32 waves = 1024 threads) per image.
//  - box table staged into LDS via CDNA5 async-to-LDS DMA (ASYNCcnt)
//  - scores + own boxes live entirely in registers (owner-writes-only)
//  - 750 iterations, 2 barriers each: wave32 shuffle argmax -> one wave
//    reduces the 32 partials -> LDS-broadcast winner -> register suppress
// ---------------------------------------------------------------------------
__global__ __launch_bounds__(1024)
void nms_kernel(const float* __restrict__ selScores,
                const float4* __restrict__ selBoxes,
                float* __restrict__ out) {
    extern __shared__ __align__(16) unsigned char smem[];
    float4* sBox = (float4*)smem;                               // K_SEL * 16B
    float*  rVal = (float*)(smem + (size_t)K_SEL * 16);         // 33 slots
    int*    rIdx = (int*)  (smem + (size_t)K_SEL * 16 + 34 * 4);

    const int img  = blockIdx.x;
    const int tid  = threadIdx.x;
    const int lane = tid & 31;
    const int wid  = tid >> 5;

    // ---- async DMA stage: global boxes -> LDS (b128 per lane) ----
    for (int j = tid; j < K_SEL; j += 1024) {
        unsigned ldsB = (unsigned)(uintptr_t)(&sBox[j]);  // flat low32 = LDS offset
        unsigned long long gB =
            (unsigned long long)(uintptr_t)(&selBoxes[(size_t)img * K_SEL + j]);
        asm volatile("global_load_async_to_lds_b128 %0, %1, off"
                     :: "v"(ldsB), "v"(gB) : "memory");
    }
    asm volatile("s_wait_asynccnt 0" ::: "memory");
    __syncthreads();

    // ---- register-resident candidate chunk (owner-only writes) ----
    float  sc[CHUNK];
    float4 bx[CHUNK];
    for (int k = 0; k < CHUNK; ++k) {
        int j = tid + k * 1024;
        if (j < K_SEL) {
            sc[k] = selScores[(size_t)img * K_SEL + j];
            bx[k] = sBox[j];
        } else {
            sc[k] = -2e9f;                 // below NEG: never beats a real entry
            bx[k] = make_float4(0.f, 0.f, 0.f, 0.f);
        }
    }

    for (int it = 0; it < TOPK; ++it) {
        // ---- argmax over registers ----
        float bv = -3e9f; int bi = 0;
        #pragma unroll
        for (int k = 0; k < CHUNK; ++k) {
            if (sc[k] > bv) { bv = sc[k]; bi = tid + k * 1024; }
        }
        #pragma unroll
        for (int off = 16; off > 0; off >>= 1) {
            float ov = __shfl_down(bv, off, 32);
            int   oi = __shfl_down(bi, off, 32);
            if (ov > bv) { bv = ov; bi = oi; }
        }
        if (lane == 0) { rVal[wid] = bv; rIdx[wid] = bi; }
        __syncthreads();                                  // B1
        if (wid == 0) {                    // 32 waves -> exactly one wave32
            float v = rVal[lane]; int i2 = rIdx[lane];
            #pragma unroll
            for (int off = 16; off > 0; off >>= 1) {
                float ov = __shfl_down(v, off, 32);
                int   oi = __shfl_down(i2, off, 32);
                if (ov > v) { v = ov; i2 = oi; }
            }
            if (lane == 0) { rVal[32] = v; rIdx[32] = i2; }
        }
        __syncthreads();                                  // B2
        const float wv = rVal[32];
        const int   wi = rIdx[32];
        const float4 wb = sBox[wi < K_SEL ? wi : 0];      // LDS broadcast

        if (tid == 0) {
            bool valid = wv > NEGF * 0.5f;
            size_t o = (((size_t)img * 2 + 1) * TOPK + it) * 5;
            out[o + 0] = valid ? wv   : 0.0f;
            out[o + 1] = valid ? wb.x : 0.0f;
            out[o + 2] = valid ? wb.y : 0.0f;
            out[o + 3] = valid ? wb.z : 0.0f;
            out[o + 4] = valid ? wb.w : 0.0f;
        }

        // ---- register suppression (no shared writes -> no extra barrier) ----
        const float wA = (wb.z - wb.x) * (wb.w - wb.y);
        #pragma unroll
        for (int k = 0; k < CHUNK; ++k) {
            float4 b = bx[k];
            float lx = fmaxf(wb.x, b.x), ly = fmaxf(wb.y, b.y);
            float rx = fminf(wb.z, b.z), ry = fminf(wb.w, b.w);
            float iw = fmaxf(rx - lx, 0.0f), ih = fmaxf(ry - ly, 0.0f);
            float inter = iw * ih;
            float uni   = wA + (b.z - b.x) * (b.w - b.y) - inter;
            float iou   = inter / fmaxf(uni, 1e-12f);
            if (iou > NMS_TH || (tid + k * 1024) == wi) sc[k] = NEGF;
        }
    }
}

// ---------------------------------------------------------------------------
extern "C" void kernel_launch(void* const* d_in, const int* in_sizes, int n_in,
                              void* d_out, int out_size, void* d_ws, size_t ws_size,
                              hipStream_t stream) {
    const float4* loc4   = (const float4*)d_in[0];   // [16,131072,4] f32
    const float2* conf2  = (const float2*)d_in[1];   // [16*131072,2] f32
    const float4* prior4 = (const float4*)d_in[2];   // [131072,4]    f32
    float* out = (float*)d_out;                      // [16,2,750,5]  f32

    char* ws = (char*)d_ws;
    unsigned* hist   = (unsigned*)(ws + 0);          // 16*256*4 = 16384
    unsigned* prefix = (unsigned*)(ws + 16384);      // 16*4
    int*      kRem   = (int*)     (ws + 16448);      // 16*4
    int*      cnt    = (int*)     (ws + 16512);      // 16*4
    float*    selScores = (float*)(ws + 16640);      // 16*5000*4 = 320000
    float4*   selBoxes  = (float4*)(ws + 336640);    // 16*5000*16 = 1280000

    init_kernel<<<(NUM_IMG * K_SEL + 255) / 256, 256, 0, stream>>>(
        out, selScores, selBoxes, prefix, kRem, cnt);

    for (int pass = 0; pass < 4; ++pass) {
        hist_kernel<<<NUM_IMG, 1024, 0, stream>>>(conf2, hist, prefix, pass);
        scan_kernel<<<1, NUM_IMG, 0, stream>>>(hist, prefix, kRem, pass);
    }

    const int total = NUM_IMG * PRIORS;
    compact_kernel<<<(total + 255) / 256, 256, 0, stream>>>(
        loc4, conf2, prior4, prefix, cnt, selScores, selBoxes, /*eq=*/0);
    compact_kernel<<<(total + 255) / 256, 256, 0, stream>>>(
        loc4, conf2, prior4, prefix, cnt, selScores, selBoxes, /*eq=*/1);

    const size_t smemBytes = (size_t)K_SEL * 16 + 34 * 4 * 2 + 64; // ~80.3 KB LDS
    nms_kernel<<<NUM_IMG, 1024, smemBytes, stream>>>(selScores, selBoxes, out);
}